// SparseHyperGraphAttentionLayer_84542136254541
// MI455X (gfx1250) — compile-verified
//
#include <hip/hip_runtime.h>
#include <hip/hip_bf16.h>

// SparseHyperGraphAttentionLayer for gfx1250 (MI455X).
//
// Pipeline:
//   (1) zero d_out
//   (2) Wh = node_embs @ W1 via V_WMMA_F32_16X16X4_F32 (f32 WMMA, full precision;
//       GEMM is memory-bound so f32 WMMA is the right ceiling)
//   (3) score[n] = dot(leakyrelu(Wh[n]), a1)  (hoisted out of the per-edge loop:
//       reference applies leakyrelu/a1 to gathered rows, which only depend on node)
//   (4) per-edge masked softmax over 8 slots + weighted accumulation of Wh rows
//       into [8,128] output via LDS block-reduce + global atomics.
//
// ws layout: Wh [N*128] f32, then score [N] f32.

typedef __attribute__((ext_vector_type(2))) float v2f;
typedef __attribute__((ext_vector_type(8))) float v8f;

#define IN_F 128
#define OUT_F 128
#define ARITY 8
#define ALPHA 0.2f
#define NEG_INF (-9e15f)

// ---------------------------------------------------------------- zero output
__global__ void shga_zero_kernel(float* __restrict__ p, int n) {
    int i = blockIdx.x * blockDim.x + threadIdx.x;
    if (i < n) p[i] = 0.0f;
}

// ---------------------------------------------------------------- GEMM (WMMA)
// One wave computes a 16-row x 128-col stripe of Wh.
// A (16x4 f32) fragment layout (ISA 7.12.2): lane<16 -> M=lane, K={0,1} in v0,v1;
// lane>=16 -> M=lane-16, K={2,3}.  => per-lane contiguous float2 at col k+2*hi.
// B (4x16 f32) fragment: v0 = W[k+2*hi][n0+lo], v1 = W[k+2*hi+1][n0+lo].
// C/D (16x16 f32): VGPR r holds row (r + 8*hi), col n0+lo.
__global__ void shga_gemm_kernel(const float* __restrict__ A,   // node_embs [N,128]
                                 const float* __restrict__ W,   // W1 [128,128]
                                 float* __restrict__ Wh,        // out [N,128]
                                 int N) {
    const int gwave = (blockIdx.x * blockDim.x + threadIdx.x) >> 5;
    const int lane  = threadIdx.x & 31;
    const int row0  = gwave * 16;
    if (row0 >= N) return;                      // wave-uniform branch (EXEC stays all-1)

    const int hi = lane >> 4;                   // 0 or 1
    const int lo = lane & 15;

    if (row0 + 16 > N) {                        // wave-uniform scalar tail (N%16!=0 safety)
        for (int r = row0; r < N; ++r)
            for (int c = lane; c < OUT_F; c += 32) {
                float s = 0.0f;
                for (int k = 0; k < IN_F; ++k)
                    s += A[(size_t)r * IN_F + k] * W[(size_t)k * OUT_F + c];
                Wh[(size_t)r * OUT_F + c] = s;
            }
        return;
    }

    v8f acc[8];
    #pragma unroll
    for (int t = 0; t < 8; ++t) { v8f z = {}; acc[t] = z; }

    const float* arow = A + (size_t)(row0 + lo) * IN_F;

    for (int k = 0; k < IN_F; k += 4) {
        // A fragment: contiguous float2 per lane
        const float2 af = *reinterpret_cast<const float2*>(arow + k + 2 * hi);
        v2f a; a.x = af.x; a.y = af.y;
        const float* wr0 = W + (size_t)(k + 2 * hi) * OUT_F;
        const float* wr1 = wr0 + OUT_F;
        #pragma unroll
        for (int nt = 0; nt < 8; ++nt) {
            const int n = nt * 16 + lo;
            v2f b; b.x = wr0[n]; b.y = wr1[n];
            acc[nt] = __builtin_amdgcn_wmma_f32_16x16x4_f32(
                false, a, false, b, (short)0, acc[nt], false, false);
        }
    }

    // store D tiles
    #pragma unroll
    for (int nt = 0; nt < 8; ++nt) {
        #pragma unroll
        for (int r = 0; r < 8; ++r) {
            Wh[(size_t)(row0 + r + 8 * hi) * OUT_F + nt * 16 + lo] = acc[nt][r];
        }
    }
}

// ---------------------------------------------------------------- node scores
// One wave per node: score[n] = sum_f leakyrelu(Wh[n][f]) * a1[f]
__global__ void shga_score_kernel(const float* __restrict__ Wh,
                                  const float* __restrict__ a1,
                                  float* __restrict__ score, int N) {
    const int gwave = (blockIdx.x * blockDim.x + threadIdx.x) >> 5;
    const int lane  = threadIdx.x & 31;
    if (gwave >= N) return;

    const float4 v = *(reinterpret_cast<const float4*>(Wh + (size_t)gwave * OUT_F) + lane);
    const float4 a = *(reinterpret_cast<const float4*>(a1) + lane);

    auto lrelu = [](float x) { return x > 0.0f ? x : ALPHA * x; };
    float p = lrelu(v.x) * a.x + lrelu(v.y) * a.y + lrelu(v.z) * a.z + lrelu(v.w) * a.w;

    #pragma unroll
    for (int off = 16; off > 0; off >>= 1) p += __shfl_xor(p, off, 32);
    if (lane == 0) score[gwave] = p;
}

// ---------------------------------------------------------------- edge reduce
// One wave per edge (grid-stride). lane l owns features 4l..4l+3 (float4).
// Per-wave register accumulator [8 slots][4 floats] -> LDS ds_add_f32 -> global atomics.
__global__ void shga_edge_kernel(const float* __restrict__ Wh,
                                 const float* __restrict__ score,
                                 const int* __restrict__ edge_list,          // [M,8]
                                 const unsigned char* __restrict__ edge_mask, // [M,8] bool
                                 float* __restrict__ out,                    // [8,128]
                                 int M) {
    __shared__ float lacc[ARITY * OUT_F];        // 4 KB
    const int tid = threadIdx.x;
    for (int i = tid; i < ARITY * OUT_F; i += blockDim.x) lacc[i] = 0.0f;
    __syncthreads();

    const int lane   = tid & 31;
    const int wid    = tid >> 5;
    const int wpb    = blockDim.x >> 5;
    const int gwave  = blockIdx.x * wpb + wid;
    const int nwaves = gridDim.x * wpb;

    float4 acc[ARITY];
    #pragma unroll
    for (int a = 0; a < ARITY; ++a) acc[a] = make_float4(0.f, 0.f, 0.f, 0.f);

    for (int e = gwave; e < M; e += nwaves) {
        // indices (32B row, broadcast across the wave)
        const int4* ep = reinterpret_cast<const int4*>(edge_list + (size_t)e * ARITY);
        const int4 i0 = ep[0], i1 = ep[1];
        int idx[ARITY] = { i0.x, i0.y, i0.z, i0.w, i1.x, i1.y, i1.z, i1.w };

        // mask (8 bytes, jax bool = 1 byte/elem)
        const uint2 mraw = *reinterpret_cast<const uint2*>(edge_mask + (size_t)e * ARITY);
        unsigned msk = 0;
        #pragma unroll
        for (int a = 0; a < 4; ++a) {
            if ((mraw.x >> (8 * a)) & 0xffu) msk |= 1u << a;
            if ((mraw.y >> (8 * a)) & 0xffu) msk |= 1u << (a + 4);
        }

        // gather scores, masked softmax over 8 slots
        float s[ARITY], mx = NEG_INF;
        #pragma unroll
        for (int a = 0; a < ARITY; ++a) {
            s[a] = ((msk >> a) & 1u) ? score[idx[a]] : NEG_INF;
            mx = fmaxf(mx, s[a]);
        }
        float w[ARITY], sum = 0.0f;
        #pragma unroll
        for (int a = 0; a < ARITY; ++a) {
            w[a] = ((msk >> a) & 1u) ? expf(s[a] - mx) : 0.0f;
            sum += w[a];
        }
        const float inv = 1.0f / sum;

        // weighted accumulation of gathered rows (coalesced: lane l -> 16B @ 16l)
        #pragma unroll
        for (int a = 0; a < ARITY; ++a) {
            if ((msk >> a) & 1u) {               // wave-uniform predicate
                const float4 g =
                    *(reinterpret_cast<const float4*>(Wh + (size_t)idx[a] * OUT_F) + lane);
                const float wa = w[a] * inv;
                acc[a].x += wa * g.x; acc[a].y += wa * g.y;
                acc[a].z += wa * g.z; acc[a].w += wa * g.w;
            }
        }
    }

    // block reduce via LDS float atomics (ds_add_f32)
    #pragma unroll
    for (int a = 0; a < ARITY; ++a) {
        atomicAdd(&lacc[a * OUT_F + 4 * lane + 0], acc[a].x);
        atomicAdd(&lacc[a * OUT_F + 4 * lane + 1], acc[a].y);
        atomicAdd(&lacc[a * OUT_F + 4 * lane + 2], acc[a].z);
        atomicAdd(&lacc[a * OUT_F + 4 * lane + 3], acc[a].w);
    }
    __syncthreads();

    for (int i = tid; i < ARITY * OUT_F; i += blockDim.x)
        atomicAdd(&out[i], lacc[i]);
}

// ---------------------------------------------------------------- launch
extern "C" void kernel_launch(void* const* d_in, const int* in_sizes, int n_in,
                              void* d_out, int out_size, void* d_ws, size_t ws_size,
                              hipStream_t stream) {
    const float*         node_embs = (const float*)d_in[0];
    // d_in[1] (edge_embs) unused by the reference math
    const int*           edge_list = (const int*)d_in[2];
    const unsigned char* edge_mask = (const unsigned char*)d_in[3];
    const float*         W1        = (const float*)d_in[4];
    const float*         a1        = (const float*)d_in[5];
    float*               out       = (float*)d_out;

    const int N = in_sizes[0] / IN_F;     // 100000 nodes
    const int M = in_sizes[2] / ARITY;    // 100000 hyperedges

    float* Wh    = (float*)d_ws;                     // N*128 f32 (51.2 MB)
    float* score = Wh + (size_t)N * OUT_F;           // N f32

    // (1) zero output accumulator
    shga_zero_kernel<<<(out_size + 255) / 256, 256, 0, stream>>>(out, out_size);

    // (2) Wh = node_embs @ W1  (WMMA f32 16x16x4; 1 wave per 16-row stripe)
    {
        const int row_tiles = (N + 15) / 16;
        const int wpb = 256 / 32;
        const int blocks = (row_tiles + wpb - 1) / wpb;
        shga_gemm_kernel<<<blocks, 256, 0, stream>>>(node_embs, W1, Wh, N);
    }

    // (3) per-node attention scores (1 wave per node)
    {
        const long long threads = (long long)N * 32;
        const int blocks = (int)((threads + 255) / 256);
        shga_score_kernel<<<blocks, 256, 0, stream>>>(Wh, a1, score, N);
    }

    // (4) per-edge softmax + weighted reduction into [8,128]
    shga_edge_kernel<<<512, 256, 0, stream>>>(Wh, score, edge_list, edge_mask, out, M);
}